// QuantumNeuralNetwork_27333171872121
// MI455X (gfx1250) — compile-verified
//
#include <hip/hip_runtime.h>

#define NQ      20
#define DIM     (1u << NQ)      // 1048576
#define NLAYERS 4
#define NBATCH  4

typedef __attribute__((ext_vector_type(2))) float v2f;
typedef __attribute__((ext_vector_type(8))) float v8f;

// ---------------------------------------------------------------------------
// Wave-level dense 16x16 (real) matrix apply via V_WMMA_F32_16X16X4_F32.
// M (row-major 16x16, precomputed in global memory) is the tensor product of
// four RY(theta/2) 2x2 matrices; bit j of the transformed nibble corresponds
// to qubit (qhi - j). X is a 16x16 tile of one real plane of the statevector
// in LDS; COLMAJOR: transformed nibble = low 4 bits (elem = col*16 + row),
// else high 4 bits (elem = row*16 + col).
// ---------------------------------------------------------------------------

template <bool COLMAJOR>
__device__ __forceinline__ v8f wave_mm16(const float* plane, const float* __restrict__ M,
                                         int lane) {
  const int l15   = lane & 15;
  const int khalf = (lane >> 4) << 1;               // lanes 16-31 hold K pair {2,3}
  v8f acc = {0.f, 0.f, 0.f, 0.f, 0.f, 0.f, 0.f, 0.f};
#pragma unroll
  for (int k = 0; k < 4; ++k) {
    const int kc = 4 * k + khalf;
    // A (16x4 chunk of M): lane = row, 2 K-columns per lane (8B aligned)
    const v2f A = *(const v2f*)(M + l15 * 16 + kc);
    v2f B;                                          // rows kc,kc+1 at column l15
    if (COLMAJOR) {
      B.x = plane[l15 * 16 + kc];
      B.y = plane[l15 * 16 + kc + 1];
    } else {
      B.x = plane[kc * 16 + l15];
      B.y = plane[(kc + 1) * 16 + l15];
    }
#if __has_builtin(__builtin_amdgcn_wmma_f32_16x16x4_f32)
    acc = __builtin_amdgcn_wmma_f32_16x16x4_f32(false, A, false, B, (short)0, acc, false, false);
#else
    asm volatile("v_wmma_f32_16x16x4_f32 %0, %1, %2, %0" : "+v"(acc) : "v"(A), "v"(B));
#endif
  }
  return acc;
}

template <bool COLMAJOR>
__device__ __forceinline__ void wave_apply16(float* plane, const float* __restrict__ M,
                                             int lane) {
  const v8f d     = wave_mm16<COLMAJOR>(plane, M, lane);
  const int l15   = lane & 15;
  const int rbase = (lane >> 4) << 3;               // C/D: VGPR v -> row v (+8 upper half)
#pragma unroll
  for (int v = 0; v < 8; ++v) {
    const int row = rbase + v;
    if (COLMAJOR) plane[l15 * 16 + row] = d[v];
    else          plane[row * 16 + l15] = d[v];
  }
}

// ---------------------------------------------------------------------------
// One-shot prep: per layer, 5 stage matrices (qhi = 19,15,11,7,3) and the
// 16-entry RZ high-nibble phase table. All RY trig lives here.
// grid = NLAYERS*5 blocks x 256 threads; mats[(layer*5+stage)*256 + r*16+c].
// ---------------------------------------------------------------------------
__global__ void __launch_bounds__(256) k_prep(const float* __restrict__ params,
                                              float* __restrict__ mats,
                                              float2* __restrict__ hitab) {
  const int layer = blockIdx.x / 5;
  const int stage = blockIdx.x % 5;
  const float* theta = params + layer * (2 * NQ);
  const int qhi = 19 - 4 * stage;
  const int e  = threadIdx.x;
  const int r  = e >> 4;
  const int cc = e & 15;
  float p = 1.0f;
#pragma unroll
  for (int j = 0; j < 4; ++j) {
    const float t  = 0.5f * theta[qhi - j];
    const float cj = cosf(t);
    const float sj = sinf(t);
    const int rb = (r >> j) & 1;
    const int cb = (cc >> j) & 1;
    p *= (rb == cb) ? cj : (rb ? sj : -sj);         // RY = [[c,-s],[s,c]]
  }
  mats[blockIdx.x * 256 + e] = p;

  if (stage == 4 && e < 16) {                       // r bit j <-> qubit 3-j
    float ang = 0.f;
#pragma unroll
    for (int j = 0; j < 4; ++j) {
      const float h = 0.5f * theta[NQ + (3 - j)];
      ang += ((e >> j) & 1) ? h : -h;
    }
    float sn, cs;
    sincosf(ang, &sn, &cs);
    hitab[layer * 16 + e] = make_float2(cs, sn);
  }
}

// ---------------------------------------------------------------------------
// Pass 1: RY on qubits 12..19 (index bits 0..7). Wave-private contiguous
// 256-amplitude tiles, two WMMA stages, in place. b128 global traffic.
// ---------------------------------------------------------------------------
__global__ void __launch_bounds__(256) k_ry_low(float2* state,
                                                const float* __restrict__ M1,
                                                const float* __restrict__ M2) {
  __shared__ float re[8][256];
  __shared__ float im[8][256];
  const int lane = threadIdx.x & 31;
  const int wave = threadIdx.x >> 5;
  const unsigned tile = blockIdx.x * 8u + wave;     // 4096 tiles
  float4* s4 = (float4*)state;                      // 2 complex per float4
  const unsigned base4 = tile * 128u;

#pragma unroll
  for (int u = 0; u < 4; ++u) {
    const float4 a = s4[base4 + u * 32 + lane];     // coalesced 512B per load
    const int e = u * 64 + lane * 2;
    re[wave][e]     = a.x;  im[wave][e]     = a.y;
    re[wave][e + 1] = a.z;  im[wave][e + 1] = a.w;
  }

  wave_apply16<true >(re[wave], M1, lane);          // e bits 3..0 <-> qubits 19..16
  wave_apply16<true >(im[wave], M1, lane);
  wave_apply16<false>(re[wave], M2, lane);          // e bits 7..4 <-> qubits 15..12
  wave_apply16<false>(im[wave], M2, lane);

#pragma unroll
  for (int u = 0; u < 4; ++u) {
    const int e = u * 64 + lane * 2;
    s4[base4 + u * 32 + lane] =
        make_float4(re[wave][e], im[wave][e], re[wave][e + 1], im[wave][e + 1]);
  }
}

// ---------------------------------------------------------------------------
// Pass 2: RY on qubits 4..11 (index bits 8..15). Block tile = 4096 amps:
// i = hi<<16 | m<<8 | lo4<<4 | j,  m in [0,256) transformed, j in [0,16).
// LDS pitch 257 avoids bank conflicts on the j-sliced layout. In place.
// ---------------------------------------------------------------------------
__global__ void __launch_bounds__(256) k_ry_mid(float2* state,
                                                const float* __restrict__ MA,
                                                const float* __restrict__ MB) {
  __shared__ float re[16 * 257];
  __shared__ float im[16 * 257];
  const int t    = threadIdx.x;
  const int lane = t & 31;
  const int wave = t >> 5;
  const unsigned base = ((blockIdx.x >> 4) << 16) | ((blockIdx.x & 15u) << 4);

#pragma unroll
  for (int u = 0; u < 16; ++u) {
    const int m = u * 16 + (t >> 4);
    const int j = t & 15;
    const float2 a = state[base + (unsigned)m * 256u + j];
    re[j * 257 + m] = a.x;
    im[j * 257 + m] = a.y;
  }
  __syncthreads();

#pragma unroll
  for (int sl = 0; sl < 2; ++sl) {
    const int j = wave * 2 + sl;                    // each wave owns 2 slices
    float* pr = &re[j * 257];
    float* pi = &im[j * 257];
    wave_apply16<true >(pr, MA, lane);              // m bits 3..0 <-> qubits 11..8
    wave_apply16<true >(pi, MA, lane);
    wave_apply16<false>(pr, MB, lane);              // m bits 7..4 <-> qubits  7..4
    wave_apply16<false>(pi, MB, lane);
  }
  __syncthreads();

#pragma unroll
  for (int u = 0; u < 16; ++u) {
    const int m = u * 16 + (t >> 4);
    const int j = t & 15;
    state[base + (unsigned)m * 256u + j] = make_float2(re[j * 257 + m], im[j * 257 + m]);
  }
}

// ---------------------------------------------------------------------------
// Pass 3: RY on qubits 0..3 (index bits 16..19) + fused RZ diagonal + fused
// CNOT-ring permutation scatter into the other buffer.
// Tile: i = r<<16 | (L0 + g), r in [0,16) transformed, g in [0,16).
// ---------------------------------------------------------------------------
__global__ void __launch_bounds__(256) k_ry_hi_rz_cnot(const float2* __restrict__ in,
                                                       float2* __restrict__ out,
                                                       const float* __restrict__ M,
                                                       const float* __restrict__ theta,
                                                       const float2* __restrict__ hitab) {
  __shared__ float re[8][256];
  __shared__ float im[8][256];
  const int lane = threadIdx.x & 31;
  const int wave = threadIdx.x >> 5;
  const unsigned tile = blockIdx.x * 8u + wave;     // 4096 tiles
  const unsigned L0   = tile * 16u;                 // low-16-bit base
  const float4* in4 = (const float4*)in;

#pragma unroll
  for (int u = 0; u < 4; ++u) {
    const int e = u * 64 + lane * 2;                // e = r*16 + g (row-major)
    const int r = e >> 4;
    const int g = e & 15;                           // even
    const float4 a = in4[((unsigned)r << 15) + tile * 8u + (g >> 1)];
    re[wave][e]     = a.x;  im[wave][e]     = a.y;
    re[wave][e + 1] = a.z;  im[wave][e + 1] = a.w;
  }

  const v8f dre = wave_mm16<false>(re[wave], M, lane);   // r bits 3..0 <-> qubits 3..0
  const v8f dim = wave_mm16<false>(im[wave], M, lane);

  const int g     = lane & 15;
  const int rbase = (lane >> 4) << 3;

  // RZ angle, low 16 bits (qubits 4..19) -- shared by this lane's 8 rows.
  const unsigned lo = L0 + g;
  float angL = 0.f;
#pragma unroll
  for (int p = 0; p < 16; ++p) {                    // index bit p <-> qubit 19-p
    const float h = 0.5f * theta[NQ + (19 - p)];
    angL += ((lo >> p) & 1u) ? h : -h;
  }
  float sL, cL;
  sincosf(angL, &sL, &cL);

#pragma unroll
  for (int v = 0; v < 8; ++v) {
    const int r = rbase + v;
    const unsigned i = ((unsigned)r << 16) + lo;
    // total phase = low * high (precomputed per 4-bit high nibble)
    const float2 hi = hitab[r];
    const float cA = cL * hi.x - sL * hi.y;
    const float sA = sL * hi.x + cL * hi.y;
    const float nr = dre[v] * cA - dim[v] * sA;
    const float ni = dre[v] * sA + dim[v] * cA;
    // CNOT ring q -> q+1 (mod 20): prefix-XOR over qubit bits (closed form).
    unsigned y = i;
    y ^= y >> 1;  y ^= y >> 2;  y ^= y >> 4;  y ^= y >> 8;  y ^= y >> 16;
    const unsigned dest = (y & 0x7ffffu) | (((y ^ (i >> 19)) & 1u) << 19);
    out[dest] = make_float2(nr, ni);
  }
}

// ---------------------------------------------------------------------------
// Pauli-Z expectations: z[q] = sum |amp|^2 * (bit_q ? -1 : +1).
// Processes amplitude pairs (b128); pair members differ only in qubit 19.
// ---------------------------------------------------------------------------
__global__ void __launch_bounds__(256) k_measure(const float2* __restrict__ state,
                                                 float* __restrict__ out20) {
  float z[NQ];
#pragma unroll
  for (int q = 0; q < NQ; ++q) z[q] = 0.f;

  const float4* s4 = (const float4*)state;
  const unsigned stride = gridDim.x * blockDim.x;
  for (unsigned ii = blockIdx.x * blockDim.x + threadIdx.x; ii < DIM / 2; ii += stride) {
    const float4 a = s4[ii];
    const float p0 = a.x * a.x + a.y * a.y;
    const float p1 = a.z * a.z + a.w * a.w;
    const float ps = p0 + p1;
    const unsigned i0 = ii * 2u;
#pragma unroll
    for (int q = 0; q < NQ - 1; ++q)
      z[q] += ((i0 >> (19 - q)) & 1u) ? -ps : ps;
    z[NQ - 1] += p0 - p1;
  }
#pragma unroll
  for (int q = 0; q < NQ; ++q) {
    float v = z[q];
#pragma unroll
    for (int off = 16; off >= 1; off >>= 1) v += __shfl_xor(v, off, 32);
    if ((threadIdx.x & 31) == 0) atomicAdd(&out20[q], v);
  }
}

__global__ void k_init_state(float2* s) {
  const unsigned i = blockIdx.x * 256u + threadIdx.x;
  s[i] = make_float2(i == 0 ? 1.f : 0.f, 0.f);
}

__global__ void k_zero(float* p, int n) {
  const int i = blockIdx.x * blockDim.x + threadIdx.x;
  if (i < n) p[i] = 0.f;
}

// ---------------------------------------------------------------------------

extern "C" void kernel_launch(void* const* d_in, const int* in_sizes, int n_in,
                              void* d_out, int out_size, void* d_ws, size_t ws_size,
                              hipStream_t stream) {
  (void)in_sizes; (void)n_in; (void)out_size; (void)ws_size;
  // d_in[0] = x (4,20) f32 -- only its batch dim matters (values unused by ref)
  const float* params = (const float*)d_in[1];      // (NLAYERS, 2*NQ) f32
  float* out = (float*)d_out;                       // (NBATCH, NQ) f32

  float2* A     = (float2*)d_ws;                    // ping-pong statevectors (2 x 8MB)
  float2* B     = A + DIM;
  float*  mats  = (float*)(B + DIM);                // NLAYERS*5*256 floats
  float2* hitab = (float2*)(mats + NLAYERS * 5 * 256);  // NLAYERS*16 float2

  k_prep<<<NLAYERS * 5, 256, 0, stream>>>(params, mats, hitab);
  k_init_state<<<DIM / 256, 256, 0, stream>>>(A);
  k_zero<<<1, 128, 0, stream>>>(out, NBATCH * NQ);

  float2* cur = A;
  float2* nxt = B;
  for (int b = 0; b < NBATCH; ++b) {
    for (int l = 0; l < NLAYERS; ++l) {
      const float* Mb = mats + (l * 5) * 256;       // stages qhi = 19,15,11,7,3
      const float* th = params + l * (2 * NQ);
      k_ry_low<<<512, 256, 0, stream>>>(cur, Mb, Mb + 256);
      k_ry_mid<<<256, 256, 0, stream>>>(cur, Mb + 2 * 256, Mb + 3 * 256);
      k_ry_hi_rz_cnot<<<512, 256, 0, stream>>>(cur, nxt, Mb + 4 * 256, th, hitab + l * 16);
      float2* t = cur; cur = nxt; nxt = t;
    }
    k_measure<<<256, 256, 0, stream>>>(cur, out + b * NQ);
  }
}